// SparseLayer_2241972928734
// MI455X (gfx1250) — compile-verified
//
#include <hip/hip_runtime.h>
#include <hip/hip_bf16.h>

typedef float v2f __attribute__((ext_vector_type(2)));
typedef float v8f __attribute__((ext_vector_type(8)));

#define N_NETS 100000
#define IFACE  5
#define BATCH  128
#define KCH    20          // K-chunks of 4: K = 80 = 16 nets * 5 inputs
#define CHUNKS 4           // batch chunks of 16 handled per wave

// ---------------------------------------------------------------------------
// Phase 1: fold the 3 linear layers of each tiny net into a single [1x5] row.
//   t[h]    = sum_g w2[g] * w1[g][h]
//   weff[i] = sum_h t[h]  * w0[h][i]
// ---------------------------------------------------------------------------
__global__ __launch_bounds__(256) void collapse_weights_kernel(
    const float* __restrict__ w0,   // [N, 2, 5]
    const float* __restrict__ w1,   // [N, 2, 2]
    const float* __restrict__ w2,   // [N, 1, 2]
    float* __restrict__ weff)       // [N, 5]
{
    int n = blockIdx.x * blockDim.x + threadIdx.x;
    if (n >= N_NETS) return;
    const float* W0 = w0 + n * 10;
    const float* W1 = w1 + n * 4;
    const float* W2 = w2 + n * 2;
    float a = W2[0], b = W2[1];
    float t0 = fmaf(a, W1[0], b * W1[2]);
    float t1 = fmaf(a, W1[1], b * W1[3]);
    float* E = weff + n * IFACE;
#pragma unroll
    for (int i = 0; i < IFACE; ++i)
        E[i] = fmaf(t0, W0[i], t1 * W0[IFACE + i]);
}

// ---------------------------------------------------------------------------
// Phase 2: grouped block-diagonal GEMM, one wave = 16 nets x 64 batch cols.
//   C[16x16] = A[16x80 block-diag of w_eff] * B[80x16 slice of x]
// A is built ONCE per wave (20 v2f fragments, all selects constant-folded)
// and reused across 4 batch chunks -> 80 WMMAs per wave.
// 12500 waves = 3125 blocks x 4 waves covers 100000 nets x 128 batch exactly,
// so EXEC is all-ones for every WMMA.
// ---------------------------------------------------------------------------
__global__ __launch_bounds__(128) void sparse_mlp_wmma_kernel(
    const float* __restrict__ x,    // [N*5, 128]
    const float* __restrict__ weff, // [N, 5]
    float* __restrict__ out)        // [N, 128]
{
    const int lane = threadIdx.x & 31;
    const int gid  = blockIdx.x * 4 + (threadIdx.x >> 5); // 0..12499
    const int grp  = gid >> 1;                            // net group 0..6249
    const int half = gid & 1;                             // batch half 0/1
    const int n0   = grp * 16;
    const int bh   = half * (CHUNKS * 16);                // 0 or 64
    const int m    = lane & 15;                           // tile row/col index
    const bool hi  = lane >= 16;

    // Per-lane w_eff row for net n0+m (lanes 16..31 mirror lanes 0..15).
    float wv[IFACE];
    {
        const float* E = weff + (size_t)(n0 + m) * IFACE;
#pragma unroll
        for (int i = 0; i < IFACE; ++i) wv[i] = E[i];
    }

    // Build the block-diagonal A fragments. All wv[] indices and all (m==g)
    // compares are compile-time constants -> pure cndmask selects, no scratch.
    v2f A[KCH];
#pragma unroll
    for (int kc = 0; kc < KCH; ++kc) {
        const int kl0 = 4 * kc,     kl1 = 4 * kc + 1;
        const int kh0 = 4 * kc + 2, kh1 = 4 * kc + 3;
        const float xl = (m == kl0 / IFACE) ? wv[kl0 % IFACE] : 0.0f;
        const float xh = (m == kh0 / IFACE) ? wv[kh0 % IFACE] : 0.0f;
        const float yl = (m == kl1 / IFACE) ? wv[kl1 % IFACE] : 0.0f;
        const float yh = (m == kh1 / IFACE) ? wv[kh1 % IFACE] : 0.0f;
        A[kc].x = hi ? xh : xl;
        A[kc].y = hi ? yh : yl;
    }

    // Per-lane x base: row (n0*5 + 2*hi), column (bh + m). All further loads
    // are immediate offsets: B vgpr0 <- row +4kc, vgpr1 <- row +4kc+1.
    const float* px = x + ((size_t)n0 * IFACE + (hi ? 2 : 0)) * BATCH + bh + m;
    // Per-lane output base: row (n0 + 8*hi), column (bh + m).
    float* po = out + ((size_t)n0 + (hi ? 8 : 0)) * BATCH + bh + m;

#pragma unroll
    for (int ch = 0; ch < CHUNKS; ++ch) {
        v8f acc0 = {};
        v8f acc1 = {};
#pragma unroll
        for (int kc = 0; kc < KCH; ++kc) {
            v2f B;
            B.x = px[(size_t)(4 * kc)     * BATCH + ch * 16];
            B.y = px[(size_t)(4 * kc + 1) * BATCH + ch * 16];
            if (kc & 1)
                acc1 = __builtin_amdgcn_wmma_f32_16x16x4_f32(
                    false, A[kc], false, B, (short)0, acc1, false, false);
            else
                acc0 = __builtin_amdgcn_wmma_f32_16x16x4_f32(
                    false, A[kc], false, B, (short)0, acc0, false, false);
        }
        // C layout: VGPR r -> M = r + (hi ? 8 : 0), col = m.
#pragma unroll
        for (int r = 0; r < 8; ++r)
            po[(size_t)r * BATCH + ch * 16] = acc0[r] + acc1[r];
    }
}

extern "C" void kernel_launch(void* const* d_in, const int* in_sizes, int n_in,
                              void* d_out, int out_size, void* d_ws, size_t ws_size,
                              hipStream_t stream) {
    const float* x  = (const float*)d_in[0];   // [N*5, 128]
    const float* w0 = (const float*)d_in[1];   // [N, 2, 5]
    const float* w1 = (const float*)d_in[2];   // [N, 2, 2]
    const float* w2 = (const float*)d_in[3];   // [N, 1, 2]
    float* out  = (float*)d_out;               // [N, 128]
    float* weff = (float*)d_ws;                // [N, 5] scratch (2 MB)

    collapse_weights_kernel<<<(N_NETS + 255) / 256, 256, 0, stream>>>(w0, w1, w2, weff);
    // 100000/16 = 6250 net groups, 2 batch halves -> 12500 waves, 4 waves/block.
    sparse_mlp_wmma_kernel<<<(N_NETS / 16) * 2 / 4, 128, 0, stream>>>(x, weff, out);
}